// AFNO2D_77360950936330
// MI455X (gfx1250) — compile-verified
//
#include <hip/hip_runtime.h>
#include <math.h>

// AFNO2D on MI455X (gfx1250): all DFT / MLP matmuls via V_WMMA_F32_16X16X4_F32.
// fp32 throughout (reference is fp32; FFT round trip punishes low precision).

typedef float v2f __attribute__((ext_vector_type(2)));
typedef float v8f __attribute__((ext_vector_type(8)));

#define B_DIM 4
#define H_DIM 128
#define W_DIM 128
#define C_DIM 768
#define WF    65      // W/2+1
#define NB    8
#define BS    96

static __device__ __forceinline__ v8f wmma_f32(v2f a, v2f b, v8f c) {
  // D = A(16x4) * B(4x16) + C(16x16), fp32
  return __builtin_amdgcn_wmma_f32_16x16x4_f32(
      /*neg_a=*/false, a, /*neg_b=*/false, b,
      /*c_mod=*/(short)0, c, /*reuse_a=*/false, /*reuse_b=*/false);
}

// ---------------------------------------------------------------------------
// DFT coefficient tables (workspace). Ortho norm 1/sqrt(H*W)=1/128 folded into
// stage1 (forward) and stage5 (inverse).
//   t1: rfft along W   [80(pad from 65) x 128]  Fr=cos*s, Fi=-sin*s
//   t2: fft along H    [128 x 128] Cr=cos, Ci=-sin, Cin=+sin (= -Ci)
//   t4: ifft along H   [128 x 128] Cr=cos, Ci=+sin, Cin=-sin
//   t5: irfft along W  [128 x 68(pad from 65)]  Gr=a*cos*s, Gi=-a*sin*s, a={1,2,..,2,1}
// ---------------------------------------------------------------------------
__global__ void init_tables(float* t1r, float* t1i,
                            float* t2r, float* t2i, float* t2in,
                            float* t4r, float* t4i, float* t4in,
                            float* t5r, float* t5i) {
  int id = blockIdx.x * blockDim.x + threadIdx.x;   // 0..16383
  const float step = 6.28318530717958647692f / 128.0f;
  const float s = 0.0078125f;                        // 1/128
  if (id < 128 * 128) {
    int hf = id / 128, h = id % 128;
    int r = (hf * h) & 127;
    float a = (float)r * step;
    float c = cosf(a), sn = sinf(a);
    t2r[id] = c;  t2i[id] = -sn; t2in[id] = sn;
    t4r[id] = c;  t4i[id] = sn;  t4in[id] = -sn;
  }
  if (id < 80 * 128) {
    int wf = id / 128, w = id % 128;
    float vc = 0.0f, vs = 0.0f;
    if (wf < WF) {
      int r = (wf * w) & 127;
      float a = (float)r * step;
      vc = cosf(a) * s; vs = -sinf(a) * s;
    }
    t1r[id] = vc; t1i[id] = vs;
  }
  if (id < 128 * 68) {
    int w = id / 68, wf = id % 68;
    float vc = 0.0f, vs = 0.0f;
    if (wf < WF) {
      float alpha = (wf == 0 || wf == 64) ? 1.0f : 2.0f;
      int r = (w * wf) & 127;
      float a = (float)r * step;
      vc = alpha * cosf(a) * s; vs = -alpha * sinf(a) * s;
    }
    t5r[id] = vc; t5i[id] = vs;
  }
}

// ---------------------------------------------------------------------------
// Stage 1: rfft along W.  Per (b,h): Y[wf,c] = F[wf,w] * x[w,c]
//   M=65 (5 tiles, padded table rows zero), K=128, N=768 (wave does 16x64)
// ---------------------------------------------------------------------------
__global__ __launch_bounds__(32)
void stage1_rfft_w(const float* __restrict__ x,
                   const float* __restrict__ t1r, const float* __restrict__ t1i,
                   float* __restrict__ yr, float* __restrict__ yi) {
  const int bh = blockIdx.x;                 // 0..511
  const int mt = blockIdx.y % 5;
  const int ng = blockIdx.y / 5;             // 0..11
  const int lane = threadIdx.x, half = lane >> 4, l16 = lane & 15;
  const int m0 = mt * 16, n0 = ng * 64;
  const float* X = x + (long)bh * W_DIM * C_DIM;
  v8f vz = {0.f,0.f,0.f,0.f,0.f,0.f,0.f,0.f};
  v8f accr[4], acci[4];
#pragma unroll
  for (int t = 0; t < 4; ++t) { accr[t] = vz; acci[t] = vz; }
  const int arow = (m0 + l16) * 128;
  for (int kk = 0; kk < 128; kk += 4) {
    const int ka = kk + 2 * half;
    v2f ar = *(const v2f*)(t1r + arow + ka);
    v2f ai = *(const v2f*)(t1i + arow + ka);
#pragma unroll
    for (int t = 0; t < 4; ++t) {
      const float* bp = X + (long)ka * C_DIM + n0 + t * 16 + l16;
      v2f bv; bv.x = bp[0]; bv.y = bp[C_DIM];
      accr[t] = wmma_f32(ar, bv, accr[t]);
      acci[t] = wmma_f32(ai, bv, acci[t]);
    }
  }
  const long obase = (long)bh * WF * C_DIM;
#pragma unroll
  for (int t = 0; t < 4; ++t)
#pragma unroll
    for (int r = 0; r < 8; ++r) {
      int wf = m0 + 8 * half + r;
      if (wf < WF) {
        long o = obase + (long)wf * C_DIM + n0 + t * 16 + l16;
        yr[o] = accr[t][r];
        yi[o] = acci[t][r];
      }
    }
}

// ---------------------------------------------------------------------------
// Stages 2 & 4: complex DFT along H.  Per (b,wf):
//   Or = tr (X) Ir + tin (X) Ii    (tin = -ti)
//   Oi = tr (X) Ii + ti  (X) Ir
//   M=128 (8 tiles), K=128, N=768
// ---------------------------------------------------------------------------
__global__ __launch_bounds__(32)
void fft_h_kernel(const float* __restrict__ inr, const float* __restrict__ ini,
                  const float* __restrict__ tr, const float* __restrict__ ti,
                  const float* __restrict__ tin,
                  float* __restrict__ outr, float* __restrict__ outi) {
  const int b = blockIdx.x / WF, wf = blockIdx.x % WF;
  const int mt = blockIdx.y & 7, ng = blockIdx.y >> 3;   // 8 x 12
  const int lane = threadIdx.x, half = lane >> 4, l16 = lane & 15;
  const int m0 = mt * 16, n0 = ng * 64;
  const long ldx = (long)WF * C_DIM;
  const long ibase = ((long)b * H_DIM * WF + wf) * C_DIM;
  v8f vz = {0.f,0.f,0.f,0.f,0.f,0.f,0.f,0.f};
  v8f accR[4], accI[4];
#pragma unroll
  for (int t = 0; t < 4; ++t) { accR[t] = vz; accI[t] = vz; }
  const int arow = (m0 + l16) * 128;
  for (int kk = 0; kk < 128; kk += 4) {
    const int ka = kk + 2 * half;
    v2f ar = *(const v2f*)(tr  + arow + ka);
    v2f an = *(const v2f*)(tin + arow + ka);
    v2f av = *(const v2f*)(ti  + arow + ka);
#pragma unroll
    for (int t = 0; t < 4; ++t) {
      const long p = ibase + (long)ka * ldx + n0 + t * 16 + l16;
      v2f br; br.x = inr[p]; br.y = inr[p + ldx];
      v2f bi; bi.x = ini[p]; bi.y = ini[p + ldx];
      accR[t] = wmma_f32(ar, br, accR[t]);
      accR[t] = wmma_f32(an, bi, accR[t]);
      accI[t] = wmma_f32(ar, bi, accI[t]);
      accI[t] = wmma_f32(av, br, accI[t]);
    }
  }
#pragma unroll
  for (int t = 0; t < 4; ++t)
#pragma unroll
    for (int r = 0; r < 8; ++r) {
      int m = m0 + 8 * half + r;
      long o = ((long)b * H_DIM * WF + (long)m * WF + wf) * C_DIM + n0 + t * 16 + l16;
      outr[o] = accR[t][r];
      outi[o] = accI[t][r];
    }
}

// ---------------------------------------------------------------------------
// Stage 3: block-diagonal complex MLP over tokens.
//   tokens T = B*H*WF = 33280; wave handles 16 tokens x 1 block (96 ch).
//   L1: o1 = GELU( x (X) W1 + b1 ) complex; L2: softshrink( o1 (X) W2 + b2 ).
// ---------------------------------------------------------------------------
__global__ __launch_bounds__(32)
void mlp_kernel(const float* __restrict__ zr, const float* __restrict__ zi,
                const float* __restrict__ w1, const float* __restrict__ b1,
                const float* __restrict__ w2, const float* __restrict__ b2,
                float* __restrict__ outr, float* __restrict__ outi) {
  __shared__ __align__(16) float sxr[16 * BS];
  __shared__ __align__(16) float sxi[16 * BS];
  __shared__ __align__(16) float so1r[16 * BS];
  __shared__ __align__(16) float so1i[16 * BS];
  const int t0 = blockIdx.x * 16;
  const int n  = blockIdx.y;
  const int lane = threadIdx.x, half = lane >> 4, l16 = lane & 15;
  for (int idx = lane; idx < 16 * BS; idx += 32) {
    int tok = idx / BS, cc = idx % BS;
    long g = (long)(t0 + tok) * C_DIM + n * BS + cc;
    sxr[idx] = zr[g];
    sxi[idx] = zi[g];
  }
  __syncthreads();
  const float* w1r = w1 + (long)n * BS * BS;
  const float* w1i = w1 + (long)(NB + n) * BS * BS;
  const float* w2r = w2 + (long)n * BS * BS;
  const float* w2i = w2 + (long)(NB + n) * BS * BS;
  const float rsqrt2 = 0.70710678118654752440f;
  v8f vz = {0.f,0.f,0.f,0.f,0.f,0.f,0.f,0.f};

  // ---- layer 1 ----
  for (int nt = 0; nt < 6; ++nt) {
    const int o0 = nt * 16;
    v8f accRp = vz, accRm = vz, accI = vz;
    for (int kk = 0; kk < BS; kk += 4) {
      const int ka = kk + 2 * half;
      v2f ar = *(const v2f*)(sxr + l16 * BS + ka);
      v2f ai = *(const v2f*)(sxi + l16 * BS + ka);
      v2f br; br.x = w1r[ka * BS + o0 + l16]; br.y = w1r[(ka + 1) * BS + o0 + l16];
      v2f bi; bi.x = w1i[ka * BS + o0 + l16]; bi.y = w1i[(ka + 1) * BS + o0 + l16];
      accRp = wmma_f32(ar, br, accRp);
      accRm = wmma_f32(ai, bi, accRm);
      accI  = wmma_f32(ai, br, accI);
      accI  = wmma_f32(ar, bi, accI);
    }
    float bre = b1[n * BS + o0 + l16];
    float bim = b1[(NB + n) * BS + o0 + l16];
#pragma unroll
    for (int r = 0; r < 8; ++r) {
      int m = 8 * half + r;
      float vr = accRp[r] - accRm[r] + bre;
      float vi = accI[r] + bim;
      vr = 0.5f * vr * (1.0f + erff(vr * rsqrt2));   // exact GELU
      vi = 0.5f * vi * (1.0f + erff(vi * rsqrt2));
      so1r[m * BS + o0 + l16] = vr;
      so1i[m * BS + o0 + l16] = vi;
    }
  }
  __syncthreads();

  // ---- layer 2 + softshrink ----
  for (int nt = 0; nt < 6; ++nt) {
    const int o0 = nt * 16;
    v8f accRp = vz, accRm = vz, accI = vz;
    for (int kk = 0; kk < BS; kk += 4) {
      const int ka = kk + 2 * half;
      v2f ar = *(const v2f*)(so1r + l16 * BS + ka);
      v2f ai = *(const v2f*)(so1i + l16 * BS + ka);
      v2f br; br.x = w2r[ka * BS + o0 + l16]; br.y = w2r[(ka + 1) * BS + o0 + l16];
      v2f bi; bi.x = w2i[ka * BS + o0 + l16]; bi.y = w2i[(ka + 1) * BS + o0 + l16];
      accRp = wmma_f32(ar, br, accRp);
      accRm = wmma_f32(ai, bi, accRm);
      accI  = wmma_f32(ai, br, accI);
      accI  = wmma_f32(ar, bi, accI);
    }
    float bre = b2[n * BS + o0 + l16];
    float bim = b2[(NB + n) * BS + o0 + l16];
#pragma unroll
    for (int r = 0; r < 8; ++r) {
      int m = 8 * half + r;
      float vr = accRp[r] - accRm[r] + bre;
      float vi = accI[r] + bim;
      vr = (vr > 0.01f) ? vr - 0.01f : ((vr < -0.01f) ? vr + 0.01f : 0.0f);
      vi = (vi > 0.01f) ? vi - 0.01f : ((vi < -0.01f) ? vi + 0.01f : 0.0f);
      long g = (long)(t0 + m) * C_DIM + n * BS + o0 + l16;
      outr[g] = vr;
      outi[g] = vi;
    }
  }
}

// ---------------------------------------------------------------------------
// Stage 5: irfft along W.  Per (b,h): out[w,c] = Gr (X) Vr + Gi (X) Vi
//   M=128 (8 tiles), K=65 padded to 68 (table zero-padded, loads guarded)
// ---------------------------------------------------------------------------
__global__ __launch_bounds__(32)
void stage5_irfft_w(const float* __restrict__ vr, const float* __restrict__ vi,
                    const float* __restrict__ t5r, const float* __restrict__ t5i,
                    float* __restrict__ out) {
  const int bh = blockIdx.x;
  const int mt = blockIdx.y & 7, ng = blockIdx.y >> 3;
  const int lane = threadIdx.x, half = lane >> 4, l16 = lane & 15;
  const int m0 = mt * 16, n0 = ng * 64;
  const long ibase = (long)bh * WF * C_DIM;
  v8f vz = {0.f,0.f,0.f,0.f,0.f,0.f,0.f,0.f};
  v8f acc[4];
#pragma unroll
  for (int t = 0; t < 4; ++t) acc[t] = vz;
  const int arow = (m0 + l16) * 68;
  for (int kk = 0; kk < 68; kk += 4) {
    const int ka = kk + 2 * half;
    v2f ar = *(const v2f*)(t5r + arow + ka);
    v2f ai = *(const v2f*)(t5i + arow + ka);
#pragma unroll
    for (int t = 0; t < 4; ++t) {
      const long p = ibase + (long)ka * C_DIM + n0 + t * 16 + l16;
      v2f br, bi;
      br.x = (ka     < WF) ? vr[p]         : 0.0f;
      br.y = (ka + 1 < WF) ? vr[p + C_DIM] : 0.0f;
      bi.x = (ka     < WF) ? vi[p]         : 0.0f;
      bi.y = (ka + 1 < WF) ? vi[p + C_DIM] : 0.0f;
      acc[t] = wmma_f32(ar, br, acc[t]);
      acc[t] = wmma_f32(ai, bi, acc[t]);
    }
  }
#pragma unroll
  for (int t = 0; t < 4; ++t)
#pragma unroll
    for (int r = 0; r < 8; ++r) {
      int w = m0 + 8 * half + r;
      out[(long)bh * W_DIM * C_DIM + (long)w * C_DIM + n0 + t * 16 + l16] = acc[t][r];
    }
}

// ---------------------------------------------------------------------------
extern "C" void kernel_launch(void* const* d_in, const int* in_sizes, int n_in,
                              void* d_out, int out_size, void* d_ws, size_t ws_size,
                              hipStream_t stream) {
  (void)in_sizes; (void)n_in; (void)out_size; (void)ws_size;
  const float* x  = (const float*)d_in[0];
  const float* w1 = (const float*)d_in[1];
  const float* b1 = (const float*)d_in[2];
  const float* w2 = (const float*)d_in[3];
  const float* b2 = (const float*)d_in[4];
  float* out = (float*)d_out;
  float* ws  = (float*)d_ws;

  const long S = (long)B_DIM * H_DIM * WF * C_DIM;   // 25,559,040 floats
  float* buf1r = ws;
  float* buf1i = ws + S;
  float* buf2r = ws + 2 * S;
  float* buf2i = ws + 3 * S;
  float* tb    = ws + 4 * S;
  float* t1r = tb;            float* t1i = tb + 10240;
  float* t2r = tb + 20480;    float* t2i = tb + 36864;   float* t2in = tb + 53248;
  float* t4r = tb + 69632;    float* t4i = tb + 86016;   float* t4in = tb + 102400;
  float* t5r = tb + 118784;   float* t5i = tb + 127488;

  init_tables<<<64, 256, 0, stream>>>(t1r, t1i, t2r, t2i, t2in,
                                      t4r, t4i, t4in, t5r, t5i);
  // forward rfft along W: x -> buf1
  stage1_rfft_w<<<dim3(B_DIM * H_DIM, 60), 32, 0, stream>>>(x, t1r, t1i, buf1r, buf1i);
  // forward fft along H: buf1 -> buf2
  fft_h_kernel<<<dim3(B_DIM * WF, 96), 32, 0, stream>>>(buf1r, buf1i, t2r, t2i, t2in,
                                                        buf2r, buf2i);
  // block-diagonal complex MLP: buf2 -> buf1
  mlp_kernel<<<dim3(2080, NB), 32, 0, stream>>>(buf2r, buf2i, w1, b1, w2, b2,
                                                buf1r, buf1i);
  // inverse fft along H: buf1 -> buf2
  fft_h_kernel<<<dim3(B_DIM * WF, 96), 32, 0, stream>>>(buf1r, buf1i, t4r, t4i, t4in,
                                                        buf2r, buf2i);
  // inverse rfft along W: buf2 -> out
  stage5_irfft_w<<<dim3(B_DIM * H_DIM, 96), 32, 0, stream>>>(buf2r, buf2i, t5r, t5i, out);
}